// HierarchicalClassifier_47777216200715
// MI455X (gfx1250) — compile-verified
//
#include <hip/hip_runtime.h>

typedef float v2f __attribute__((ext_vector_type(2)));
typedef float v8f __attribute__((ext_vector_type(8)));

#define DD 1024
#define PP 64
#define NC0 16
#define NC1 32

// ---------------------------------------------------------------------------
// Kernel 1: parent_logits[B,64] = x[B,1024] @ parent_w[64,1024]^T + parent_b
// fp32 WMMA 16x16x4. One wave per 16x16 output tile; block = 8 waves covering
// a 32-row x 64-col tile. K stepped by 4 per wmma.
// A frag (16x4, 2 VGPR/lane): lanes 0-15 -> M=lane, K={k,k+1};
//                             lanes 16-31 -> M=lane-16, K={k+2,k+3}.
// B frag (4x16) mirrors A with N in place of M (B[k][n] = parent_w[n][k]).
// C/D layout: VGPR r -> row (half*8 + r), col = lane & 15.
// ---------------------------------------------------------------------------
__global__ __launch_bounds__(256) void parent_gemm_wmma(
    const float* __restrict__ x, const float* __restrict__ pw,
    const float* __restrict__ pb, float* __restrict__ logits)
{
  const int lane = threadIdx.x & 31;
  const int wave = threadIdx.x >> 5;
  const int half = lane >> 4;   // which K-pair this lane carries
  const int idx  = lane & 15;   // M index for A, N index for B/C

  const int m0 = blockIdx.x * 32 + (wave >> 2) * 16;
  const int n0 = (wave & 3) * 16;

  const float* arow = x  + (size_t)(m0 + idx) * DD + half * 2;
  const float* brow = pw + (size_t)(n0 + idx) * DD + half * 2;

  v8f acc = {};
#pragma unroll 4
  for (int k = 0; k < DD; k += 4) {
    v2f a = *(const v2f*)(arow + k);
    v2f b = *(const v2f*)(brow + k);
    acc = __builtin_amdgcn_wmma_f32_16x16x4_f32(
        /*neg_a=*/false, a, /*neg_b=*/false, b,
        /*c_mod=*/(short)0, acc, /*reuse_a=*/false, /*reuse_b=*/false);
  }

  const float bias  = pb[n0 + idx];
  const int   mbase = m0 + half * 8;
#pragma unroll
  for (int r = 0; r < 8; ++r)
    logits[(size_t)(mbase + r) * PP + n0 + idx] = acc[r] + bias;
}

// ---------------------------------------------------------------------------
// Kernel 2: top-1 routing. argmax(softmax(z)) == argmax(z); first-max ties
// to match jnp.argmax.
// ---------------------------------------------------------------------------
__global__ __launch_bounds__(256) void parent_argmax(
    const float* __restrict__ logits, int* __restrict__ cls, int B)
{
  const int b = blockIdx.x * blockDim.x + threadIdx.x;
  if (b >= B) return;
  const float* row = logits + (size_t)b * PP;
  float best = row[0];
  int   bi   = 0;
#pragma unroll
  for (int j = 1; j < PP; ++j) {
    float v = row[j];
    if (v > best) { best = v; bi = j; }
  }
  cls[b] = bi;
}

// ---------------------------------------------------------------------------
// Kernel 3: gathered child heads. One block per sample; x row staged in LDS;
// 8 waves each compute 6 of the 48 dot products with coalesced float4 weight
// loads (L2-resident) and a wave32 xor-shuffle reduction.
// ---------------------------------------------------------------------------
__global__ __launch_bounds__(256) void child_heads(
    const float* __restrict__ x,  const int* __restrict__ cls,
    const float* __restrict__ w0, const float* __restrict__ b0,
    const float* __restrict__ w1, const float* __restrict__ b1,
    float* __restrict__ c0, float* __restrict__ c1)
{
  __shared__ float xs[DD];
  const int b = blockIdx.x;
  const int t = threadIdx.x;

  const float* xrow = x + (size_t)b * DD;
#pragma unroll
  for (int i = 0; i < DD / 256; ++i)
    xs[t + i * 256] = xrow[t + i * 256];
  __syncthreads();

  const int p    = cls[b];
  const int lane = t & 31;
  const int wave = t >> 5;

  for (int o = wave; o < NC0 + NC1; o += 8) {
    const float* wrow;
    float bias;
    float* outp;
    if (o < NC0) {
      wrow = w0 + ((size_t)p * NC0 + o) * DD;
      bias = b0[p * NC0 + o];
      outp = c0 + (size_t)b * NC0 + o;
    } else {
      const int oo = o - NC0;
      wrow = w1 + ((size_t)p * NC1 + oo) * DD;
      bias = b1[p * NC1 + oo];
      outp = c1 + (size_t)b * NC1 + oo;
    }
    float partial = 0.f;
#pragma unroll
    for (int it = 0; it < DD / 128; ++it) {
      const int j = it * 128 + lane * 4;         // contiguous 512B per wave
      const float4 wv = *(const float4*)(wrow + j);
      const float4 xv = *(const float4*)(xs + j);
      partial = fmaf(xv.x, wv.x, partial);
      partial = fmaf(xv.y, wv.y, partial);
      partial = fmaf(xv.z, wv.z, partial);
      partial = fmaf(xv.w, wv.w, partial);
    }
#pragma unroll
    for (int off = 16; off > 0; off >>= 1)
      partial += __shfl_xor(partial, off, 32);   // wave32 reduction
    if (lane == 0) *outp = partial + bias;
  }
}

// ---------------------------------------------------------------------------
extern "C" void kernel_launch(void* const* d_in, const int* in_sizes, int n_in,
                              void* d_out, int out_size, void* d_ws, size_t ws_size,
                              hipStream_t stream)
{
  const float* x  = (const float*)d_in[0];
  const float* pw = (const float*)d_in[1];
  const float* pb = (const float*)d_in[2];
  const float* w0 = (const float*)d_in[3];
  const float* b0 = (const float*)d_in[4];
  const float* w1 = (const float*)d_in[5];
  const float* b1 = (const float*)d_in[6];

  const int B = in_sizes[0] / DD;   // 4096

  float* out    = (float*)d_out;
  float* logits = out;                               // [B, 64]
  float* c0     = logits + (size_t)B * PP;           // [B, 16]
  float* c1     = c0 + (size_t)B * NC0;              // [B, 32]
  int*   clsbuf = (int*)d_ws;                        // [B] routing indices

  parent_gemm_wmma<<<B / 32, 256, 0, stream>>>(x, pw, pb, logits);
  parent_argmax  <<<(B + 255) / 256, 256, 0, stream>>>(logits, clsbuf, B);
  child_heads    <<<B, 256, 0, stream>>>(x, clsbuf, w0, b0, w1, b1, c0, c1);
}